// DistributionBasedCrossAttention_38104949850333
// MI455X (gfx1250) — compile-verified
//
#include <hip/hip_runtime.h>
#include <math.h>

// Problem constants (match reference)
#define NB 128      // batch
#define LT 64       // template len
#define LC 400      // search len
#define LL 464      // LT+LC
#define LCH 116     // rows per L-chunk (LL/4)
#define DD 768      // dim
#define HH 3072     // mlp hidden
#define MM 384      // NB*3 rows through the MLP
#define NEGV (-1e20f)

typedef __attribute__((ext_vector_type(2))) float v2f;
typedef __attribute__((ext_vector_type(8))) float v8f;

// ---------------------------------------------------------------------------
// K0: 1/max(||cls||,eps) per batch
// ---------------------------------------------------------------------------
__global__ void k_cls_norm(const float* __restrict__ cls, float* __restrict__ rn) {
    __shared__ float red[256];
    int b = blockIdx.x, t = threadIdx.x;
    float s = 0.f;
    for (int d = t; d < DD; d += 256) { float v = cls[(size_t)b * DD + d]; s += v * v; }
    red[t] = s; __syncthreads();
    for (int o = 128; o > 0; o >>= 1) { if (t < o) red[t] += red[t + o]; __syncthreads(); }
    if (t == 0) rn[b] = 1.f / fmaxf(sqrtf(red[0]), 1e-12f);
}

// ---------------------------------------------------------------------------
// K1: sim_logit[b,l] = cos(cls, tgt_row) * exp(logit_scale).
// One wave32 per row; float4 loads -> global_load_b128 (HBM-bound leg #1).
// ---------------------------------------------------------------------------
__global__ void k_sim(const float* __restrict__ tem, const float* __restrict__ ctx,
                      const float* __restrict__ cls, const float* __restrict__ rn,
                      const float* __restrict__ logit_scale, float* __restrict__ sim) {
    int b = blockIdx.y;
    int wave = threadIdx.x >> 5, lane = threadIdx.x & 31;
    int l = blockIdx.x * 8 + wave;
    if (l >= LL) return;
    const float* row = (l < LT) ? (tem + ((size_t)b * LT + l) * DD)
                                : (ctx + ((size_t)b * LC + (l - LT)) * DD);
    const float* c = cls + (size_t)b * DD;
    float dot = 0.f, nrm = 0.f;
    for (int d0 = lane * 4; d0 < DD; d0 += 128) {
        float4 x = *(const float4*)(row + d0);
        float4 cc = *(const float4*)(c + d0);
        dot += x.x * cc.x + x.y * cc.y + x.z * cc.z + x.w * cc.w;
        nrm += x.x * x.x + x.y * x.y + x.z * x.z + x.w * x.w;
    }
    for (int o = 16; o > 0; o >>= 1) {
        dot += __shfl_down(dot, o, 32);
        nrm += __shfl_down(nrm, o, 32);
    }
    if (lane == 0)
        sim[(size_t)b * LL + l] = dot * rn[b] / fmaxf(sqrtf(nrm), 1e-12f) * expf(logit_scale[0]);
}

// ---------------------------------------------------------------------------
// K2: per-batch scoring. 512-thread block: softmaxes + bitonic sort +
// inclusive scan to find the 0.25-cumsum threshold (divide_background).
// ---------------------------------------------------------------------------
__device__ __forceinline__ float blk_max(float v, float* red, int t) {
    red[t] = v; __syncthreads();
    for (int o = 256; o > 0; o >>= 1) { if (t < o) red[t] = fmaxf(red[t], red[t + o]); __syncthreads(); }
    float r = red[0]; __syncthreads(); return r;
}
__device__ __forceinline__ float blk_sum(float v, float* red, int t) {
    red[t] = v; __syncthreads();
    for (int o = 256; o > 0; o >>= 1) { if (t < o) red[t] += red[t + o]; __syncthreads(); }
    float r = red[0]; __syncthreads(); return r;
}
__device__ __forceinline__ float blk_min(float v, float* red, int t) {
    red[t] = v; __syncthreads();
    for (int o = 256; o > 0; o >>= 1) { if (t < o) red[t] = fminf(red[t], red[t + o]); __syncthreads(); }
    float r = red[0]; __syncthreads(); return r;
}

__global__ void __launch_bounds__(512)
k_score(const float* __restrict__ sim, const unsigned char* __restrict__ tmask,
        const unsigned char* __restrict__ cmask, float* __restrict__ scores) {
    __shared__ float s_sort[512];
    __shared__ float s_scan[512];
    __shared__ float s_red[512];
    int b = blockIdx.x, i = threadIdx.x;
    bool act = (i < LL);
    float sv = act ? sim[(size_t)b * LL + i] : 0.f;
    bool m = false;
    if (act) m = (i < LT) ? (tmask[b * LT + i] != 0) : (cmask[b * LC + (i - LT)] != 0);

    // tgt_score = softmax(where(mask, sim, NEG))
    float tl = act ? (m ? sv : NEGV) : -INFINITY;
    float mx = blk_max(tl, s_red, i);
    float e  = act ? expf(tl - mx) : 0.f;
    float sm = blk_sum(e, s_red, i);
    float tgt_s = e / sm;

    // bgd_score0 = softmax(where(mask, NEG, sim))
    float bl = act ? (m ? NEGV : sv) : -INFINITY;
    mx = blk_max(bl, s_red, i);
    e  = act ? expf(bl - mx) : 0.f;
    sm = blk_sum(e, s_red, i);
    float b0 = e / sm;

    // sort ascending (pads -> +inf go to the end)
    s_sort[i] = act ? b0 : INFINITY;
    __syncthreads();
    for (int k = 2; k <= 512; k <<= 1)
        for (int j = k >> 1; j > 0; j >>= 1) {
            int ixj = i ^ j;
            if (ixj > i) {
                float a = s_sort[i], bb = s_sort[ixj];
                bool up = ((i & k) == 0);
                if ((a > bb) == up) { s_sort[i] = bb; s_sort[ixj] = a; }
            }
            __syncthreads();
        }
    // inclusive scan
    s_scan[i] = s_sort[i];
    __syncthreads();
    for (int off = 1; off < 512; off <<= 1) {
        float t = (i >= off) ? s_scan[i - off] : 0.f;
        __syncthreads();
        s_scan[i] += t;
        __syncthreads();
    }
    // threshold = smallest sorted value whose inclusive cumsum >= 0.25
    float cand = (s_scan[i] >= 0.25f) ? s_sort[i] : INFINITY;
    float thr = blk_min(cand, s_red, i);
    bool dis = act && (b0 >= thr);

    // bgd_score = softmax(where(dis, NEG, bgd_logit))
    float bl2 = act ? (dis ? NEGV : bl) : -INFINITY;
    mx = blk_max(bl2, s_red, i);
    e  = act ? expf(bl2 - mx) : 0.f;
    sm = blk_sum(e, s_red, i);
    float bgd_s = e / sm;

    // dis_score = softmax(where(dis, bgd_logit, NEG))
    float dl = act ? (dis ? bl : NEGV) : -INFINITY;
    mx = blk_max(dl, s_red, i);
    e  = act ? expf(dl - mx) : 0.f;
    sm = blk_sum(e, s_red, i);
    float dis_s = e / sm;

    if (act) {
        scores[((size_t)b * 3 + 0) * LL + i] = tgt_s;
        scores[((size_t)b * 3 + 1) * LL + i] = dis_s;
        scores[((size_t)b * 3 + 2) * LL + i] = bgd_s;
    }
}

// ---------------------------------------------------------------------------
// K3a: token partial sums. Streams tgt exactly once (HBM/L2-bound leg #2).
// Each thread owns 4 consecutive d (float4 loads), each block one L-chunk.
// part[b][lc][j][d]  (deterministic two-stage reduction, no atomics)
// ---------------------------------------------------------------------------
__global__ void __launch_bounds__(64)
k_tok_part(const float* __restrict__ tem, const float* __restrict__ ctx,
           const float* __restrict__ scores, float* __restrict__ part) {
    __shared__ float sw[3 * LCH];
    int b = blockIdx.z, lc = blockIdx.y;
    int l0 = lc * LCH;
    for (int idx = threadIdx.x; idx < 3 * LCH; idx += 64)
        sw[idx] = scores[((size_t)b * 3 + idx / LCH) * LL + l0 + (idx % LCH)];
    __syncthreads();
    int d0 = blockIdx.x * 256 + threadIdx.x * 4;
    float4 a0 = {0.f, 0.f, 0.f, 0.f}, a1 = a0, a2 = a0;
    for (int r = 0; r < LCH; ++r) {
        int l = l0 + r;
        const float* row = (l < LT) ? (tem + ((size_t)b * LT + l) * DD)
                                    : (ctx + ((size_t)b * LC + (l - LT)) * DD);
        float4 x = *(const float4*)(row + d0);
        float w0 = sw[r], w1 = sw[LCH + r], w2 = sw[2 * LCH + r];
        a0.x += w0 * x.x; a0.y += w0 * x.y; a0.z += w0 * x.z; a0.w += w0 * x.w;
        a1.x += w1 * x.x; a1.y += w1 * x.y; a1.z += w1 * x.z; a1.w += w1 * x.w;
        a2.x += w2 * x.x; a2.y += w2 * x.y; a2.z += w2 * x.z; a2.w += w2 * x.w;
    }
    float* p = part + ((size_t)(b * 4 + lc) * 3) * DD + d0;
    *(float4*)(p)          = a0;
    *(float4*)(p + DD)     = a1;
    *(float4*)(p + 2 * DD) = a2;
}

// ---------------------------------------------------------------------------
// K3b: reduce 4 partials, add query_embed (+ cls for j==0) -> X [MM, DD]
// ---------------------------------------------------------------------------
__global__ void __launch_bounds__(256)
k_tok_reduce(const float* __restrict__ part, const float* __restrict__ qe,
             const float* __restrict__ cls, float* __restrict__ X) {
    int b = blockIdx.y;
    int d = blockIdx.x * 256 + threadIdx.x;
    for (int j = 0; j < 3; ++j) {
        float s = 0.f;
        for (int lc = 0; lc < 4; ++lc)
            s += part[((size_t)(b * 4 + lc) * 3 + j) * DD + d];
        s += qe[(size_t)j * DD + d];
        if (j == 0) s += cls[(size_t)b * DD + d];
        X[((size_t)b * 3 + j) * DD + d] = s;
    }
}

// ---------------------------------------------------------------------------
// K4: fc1 + bias + exact GELU via V_WMMA_F32_16X16X4_F32.
// Block = 8 waves; each wave owns one 16x16 output tile. A tile (16x768)
// staged in LDS (48KB, contiguous float4 copy) shared across 8 N-subtiles.
// A frag (ISA 7.12.2, 32-bit 16x4): lane<16 -> K{0,1}, lane>=16 -> K{2,3}.
// ---------------------------------------------------------------------------
__global__ void __launch_bounds__(256)
k_fc1(const float* __restrict__ X, const float* __restrict__ W1,
      const float* __restrict__ b1, float* __restrict__ Hm) {
    __shared__ float sA[16 * DD];
    int m0 = blockIdx.y * 16;
    {   // 16 full consecutive rows of X are one contiguous block
        const float4* src = (const float4*)(X + (size_t)m0 * DD);
        float4* dst = (float4*)sA;
        for (int i = threadIdx.x; i < 16 * DD / 4; i += 256) dst[i] = src[i];
    }
    __syncthreads();
    int wave = threadIdx.x >> 5, lane = threadIdx.x & 31;
    int half = lane >> 4, lr = lane & 15;
    int col = (blockIdx.x * 8 + wave) * 16 + lr;
    v8f c = {0.f, 0.f, 0.f, 0.f, 0.f, 0.f, 0.f, 0.f};
    for (int k0 = 0; k0 < DD; k0 += 4) {
        int ka = k0 + (half ? 2 : 0);
        v2f a, bb;
        a.x  = sA[lr * DD + ka];
        a.y  = sA[lr * DD + ka + 1];
        bb.x = W1[(size_t)ka * HH + col];
        bb.y = W1[(size_t)(ka + 1) * HH + col];
        c = __builtin_amdgcn_wmma_f32_16x16x4_f32(false, a, false, bb, (short)0, c, false, false);
    }
    float bias = b1[col];
    for (int i2 = 0; i2 < 8; ++i2) {
        int m = m0 + i2 + half * 8;
        float v = c[i2] + bias;
        v = 0.5f * v * (1.f + erff(v * 0.70710678118654752f));  // exact GELU
        Hm[(size_t)m * HH + col] = v;
    }
}

// ---------------------------------------------------------------------------
// K5: fc2 + bias + residual + per-sample flag select (fused epilogue).
// H tile staged in 512-wide K chunks (32KB LDS, float4 copies per row).
// out[b,j,:] = flag[b]==1 ? src_ : (H@W2 + b2 + X)
// ---------------------------------------------------------------------------
__global__ void __launch_bounds__(256)
k_fc2(const float* __restrict__ Hm, const float* __restrict__ W2,
      const float* __restrict__ b2, const float* __restrict__ X,
      const float* __restrict__ qe, const float* __restrict__ cls,
      const int* __restrict__ flag, float* __restrict__ out) {
    __shared__ float sH[16 * 512];
    int m0 = blockIdx.y * 16;
    int wave = threadIdx.x >> 5, lane = threadIdx.x & 31;
    int half = lane >> 4, lr = lane & 15;
    int col = (blockIdx.x * 8 + wave) * 16 + lr;
    v8f c = {0.f, 0.f, 0.f, 0.f, 0.f, 0.f, 0.f, 0.f};
    for (int kc = 0; kc < HH; kc += 512) {
        __syncthreads();
        for (int i4 = threadIdx.x; i4 < 16 * 512 / 4; i4 += 256) {
            int row = i4 >> 7;            // 128 float4 per 512-wide row
            int coff = (i4 & 127) * 4;
            *(float4*)(sH + row * 512 + coff) =
                *(const float4*)(Hm + (size_t)(m0 + row) * HH + kc + coff);
        }
        __syncthreads();
        for (int k0 = 0; k0 < 512; k0 += 4) {
            int ka = k0 + (half ? 2 : 0);
            int kg = kc + ka;
            v2f a, bb;
            a.x  = sH[lr * 512 + ka];
            a.y  = sH[lr * 512 + ka + 1];
            bb.x = W2[(size_t)kg * DD + col];
            bb.y = W2[(size_t)(kg + 1) * DD + col];
            c = __builtin_amdgcn_wmma_f32_16x16x4_f32(false, a, false, bb, (short)0, c, false, false);
        }
    }
    float bias = b2[col];
    for (int i2 = 0; i2 < 8; ++i2) {
        int m = m0 + i2 + half * 8;
        int b = m / 3, j = m % 3;
        float v = c[i2] + bias + X[(size_t)m * DD + col];
        float srcu = qe[(size_t)j * DD + col] + (j == 0 ? cls[(size_t)b * DD + col] : 0.f);
        out[(size_t)m * DD + col] = (flag[b] == 1) ? srcu : v;
    }
}

// ---------------------------------------------------------------------------
// Launch
// ---------------------------------------------------------------------------
extern "C" void kernel_launch(void* const* d_in, const int* in_sizes, int n_in,
                              void* d_out, int out_size, void* d_ws, size_t ws_size,
                              hipStream_t stream) {
    const float*         tem   = (const float*)d_in[0];
    const unsigned char* tmask = (const unsigned char*)d_in[1];  // jax bool = 1 byte
    const float*         ctx   = (const float*)d_in[2];
    const unsigned char* cmask = (const unsigned char*)d_in[3];
    const float*         cls   = (const float*)d_in[4];
    const int*           flag  = (const int*)d_in[5];
    const float*         qe    = (const float*)d_in[6];
    const float*         lsc   = (const float*)d_in[7];
    const float*         W1    = (const float*)d_in[8];
    const float*         b1    = (const float*)d_in[9];
    const float*         W2    = (const float*)d_in[10];
    const float*         b2    = (const float*)d_in[11];
    float* out = (float*)d_out;

    // workspace layout (floats): ~11.6 MB total
    float* ws     = (float*)d_ws;
    float* rn     = ws;                               // NB
    float* sim    = rn + NB;                          // NB*LL
    float* scores = sim + (size_t)NB * LL;            // NB*3*LL
    float* part   = scores + (size_t)NB * 3 * LL;     // NB*4*3*DD
    float* X      = part + (size_t)NB * 4 * 3 * DD;   // MM*DD
    float* Hm     = X + (size_t)MM * DD;              // MM*HH

    k_cls_norm<<<NB, 256, 0, stream>>>(cls, rn);
    k_sim<<<dim3((LL + 7) / 8, NB), 256, 0, stream>>>(tem, ctx, cls, rn, lsc, sim);
    k_score<<<NB, 512, 0, stream>>>(sim, tmask, cmask, scores);
    k_tok_part<<<dim3(DD / 256, 4, NB), 64, 0, stream>>>(tem, ctx, scores, part);
    k_tok_reduce<<<dim3(DD / 256, NB), 256, 0, stream>>>(part, qe, cls, X);
    k_fc1<<<dim3(HH / 16 / 8, MM / 16), 256, 0, stream>>>(X, W1, b1, Hm);
    k_fc2<<<dim3(DD / 16 / 8, MM / 16), 256, 0, stream>>>(Hm, W2, b2, X, qe, cls, flag, out);
}